// ScannedLSTM_15479062134959
// MI455X (gfx1250) — compile-verified
//
#include <hip/hip_runtime.h>

// ---------------------------------------------------------------------------
// Scanned LSTM for MI455X (gfx1250), fp32 via V_WMMA_F32_16X16X4_F32.
//   Kernel 1: Zpre = xs @ Wi + b   (GEMM, double-buffered async-LDS staging)
//   Kernel 2: sequential scan; Wh column-slice resident in LDS via TDM,
//             per-step h staging via global_load_async_to_lds_b32,
//             Zpre overlapped with the WMMA loop + prefetch of step t+1.
// ---------------------------------------------------------------------------

typedef __attribute__((ext_vector_type(2))) float v2f;
typedef __attribute__((ext_vector_type(4))) float v4f;
typedef __attribute__((ext_vector_type(8))) float v8f;

#define T_STEPS 512
#define BATCH   64
#define HID     512
#define GATES   2048   // 4*HID

#if defined(__has_builtin)
#  if __has_builtin(__builtin_amdgcn_tensor_load_to_lds)
#    define USE_TDM 1
#  endif
#endif
#ifndef USE_TDM
#  define USE_TDM 0
#endif

__device__ inline v8f vzero8() {
  v8f v = {0.f,0.f,0.f,0.f,0.f,0.f,0.f,0.f};
  return v;
}

// D = A(16x4) * B(4x16) + C, fp32 WMMA.
__device__ inline v8f wmma_f32(v2f a, v2f b, v8f c) {
  return __builtin_amdgcn_wmma_f32_16x16x4_f32(
      false, a, false, b, (short)0, c, false, false);
}

// Fragment load from LDS tile stored [k][col], row pitch `pitch` (floats).
// 32-bit 16x4 layout: lanes 0-15: col=lane, VGPR0/1 -> k0,k0+1;
//                     lanes 16-31: col=lane-16, VGPR0/1 -> k0+2,k0+3.
__device__ inline v2f load_kfrag(const float* s, int k0, int col0, int pitch) {
  int lane = threadIdx.x & 31;
  int half = lane >> 4;
  int lc   = lane & 15;
  const float* p = s + (k0 + 2 * half) * pitch + col0 + lc;
  v2f r;
  r.x = p[0];
  r.y = p[pitch];
  return r;
}

// Async copy: 4 bytes per lane, global -> LDS, per-lane LDS address (ASYNCcnt).
__device__ inline void async_g2l_b32(unsigned lds_byte_off, const void* gaddr) {
  unsigned long long ga = (unsigned long long)(uintptr_t)gaddr;
  asm volatile("global_load_async_to_lds_b32 %0, %1, off"
               :: "v"(lds_byte_off), "v"(ga) : "memory");
}
__device__ inline void wait_asynccnt0()  { asm volatile("s_wait_asynccnt 0x0"  ::: "memory"); }
__device__ inline void wait_asynccnt24() { asm volatile("s_wait_asynccnt 0x18" ::: "memory"); }

#if USE_TDM
typedef __attribute__((ext_vector_type(4))) unsigned u32x4;
typedef __attribute__((ext_vector_type(8))) int      i32x8;
typedef __attribute__((ext_vector_type(4))) int      i32x4;

// TDM: load a 32-float-wide x 512-row column strip of Wh into LDS with a
// 129-dword LDS row pitch (pad_interval=32 dwords, pad_amount=97 dwords).
__device__ inline void tdm_load_strip(const float* gsrc, unsigned lds_byte_off) {
  unsigned long long ga = (unsigned long long)(uintptr_t)gsrc;
  u32x4 g0;
  g0.x = 1u;                                            // count=1 (valid, user)
  g0.y = lds_byte_off;                                  // lds_addr (bytes)
  g0.z = (unsigned)(ga & 0xFFFFFFFFu);                  // global_addr[31:0]
  g0.w = (unsigned)((ga >> 32) & 0x1FFFFFFu)            // global_addr[56:32]
         | (2u << 30);                                  // type=2 ("image")
  i32x8 g1;
  g1[0] = (int)((2u << 16)        // data_size = 4B
              | (1u << 20)        // pad_enable
              | (4u << 22)        // pad_interval: 2<<4 = 32 dwords
              | (96u << 25));     // pad_amount: 96+1 = 97 dwords
  g1[1] = (int)(2048u << 16);     // tensor_dim0[15:0] @ bits 63:48
  g1[2] = (int)(512u << 16);      // tensor_dim1[15:0] @ bits 95:80
  g1[3] = (int)(32u << 16);       // tile_dim0 = 32     @ bits 127:112
  g1[4] = (int)512;               // tile_dim1 = 512    @ bits 143:128
  g1[5] = (int)2048;              // tensor_dim0_stride[31:0] (elements)
  g1[6] = 0;
  g1[7] = 0;
  i32x4 z4 = {0, 0, 0, 0};
#if __clang_major__ >= 23
  i32x8 z8 = {0, 0, 0, 0, 0, 0, 0, 0};
  __builtin_amdgcn_tensor_load_to_lds(g0, g1, z4, z4, z8, 0);
#else
  __builtin_amdgcn_tensor_load_to_lds(g0, g1, z4, z4, 0);
#endif
}
#endif  // USE_TDM

// ---------------------------------------------------------------------------
// Kernel 1: Zpre[m,n] = xs[m,:] @ Wi[:,n] + b[n];  M=32768, N=2048, K=512.
// 64x128 C-tile / WG, 8 waves, each wave 16x64 (4 WMMA tiles).
// Double-buffered async staging: prefetch chunk i+1 while WMMAing chunk i.
// ---------------------------------------------------------------------------
#define K1_PA 65    // A tile pitch ([k][m], 32x64 used)
#define K1_PB 129   // B tile pitch ([k][n], 32x128 used)
#define K1_BS_OFF   (32 * K1_PA * 4)                  // Bs offset within buffer
#define K1_BUF_FLOATS (32 * K1_PA + 32 * K1_PB)       // 6208 floats / buffer
#define K1_BUF_BYTES  (K1_BUF_FLOATS * 4)             // 24832 B
#define K1_LDS_BYTES  (2 * K1_BUF_BYTES)              // 49664 B

// Stage one 64x32 A chunk (transposed) + 32x128 B chunk: 24 async ops / wave.
__device__ inline void k1_stage(const float* xs, const float* Wi,
                                int m0, int n0, int kc, unsigned buf_byte,
                                int wave, int lane) {
#pragma unroll
  for (int i = 0; i < 8; ++i) {
    int m = i * 8 + wave;
    async_g2l_b32(buf_byte + (unsigned)((lane * K1_PA + m) * 4),
                  xs + (size_t)(m0 + m) * HID + kc + lane);
  }
#pragma unroll
  for (int j = 0; j < 16; ++j) {
    int idx = wave * 16 + j;       // 0..127
    int k   = idx >> 2;
    int ch  = (idx & 3) * 32;
    async_g2l_b32(buf_byte + (unsigned)K1_BS_OFF +
                      (unsigned)((k * K1_PB + ch + lane) * 4),
                  Wi + (size_t)(kc + k) * GATES + n0 + ch + lane);
  }
}

__global__ __launch_bounds__(256) void lstm_pregemm(
    const float* __restrict__ xs, const float* __restrict__ Wi,
    const float* __restrict__ bias, float* __restrict__ Z) {
  extern __shared__ float smem1[];

  const int m0   = blockIdx.x * 64;
  const int n0   = blockIdx.y * 128;
  const int tid  = threadIdx.x;
  const int wave = tid >> 5;
  const int lane = tid & 31;
  const int mstrip = (wave >> 1) * 16;
  const int nhalf  = (wave & 1) * 64;

  v8f acc[4];
  acc[0] = vzero8(); acc[1] = vzero8(); acc[2] = vzero8(); acc[3] = vzero8();

  // prologue: stage chunk 0 into buffer 0
  k1_stage(xs, Wi, m0, n0, 0, 0u, wave, lane);

  for (int ci = 0; ci < 16; ++ci) {
    const unsigned curbuf = (unsigned)(ci & 1) * K1_BUF_BYTES;
    const float* As = smem1 + (ci & 1) * K1_BUF_FLOATS;
    const float* Bs = As + 32 * K1_PA;

    if (ci < 15) {
      // prefetch chunk ci+1 into the other buffer (24 async ops / wave),
      // then wait for chunk ci only (<=24 newer ops outstanding; in-order).
      k1_stage(xs, Wi, m0, n0, (ci + 1) * 32,
               (unsigned)((ci + 1) & 1) * K1_BUF_BYTES, wave, lane);
      wait_asynccnt24();
    } else {
      wait_asynccnt0();
    }
    __syncthreads();

#pragma unroll
    for (int k0 = 0; k0 < 32; k0 += 4) {
      v2f af = load_kfrag(As, k0, mstrip, K1_PA);
#pragma unroll
      for (int j = 0; j < 4; ++j) {
        v2f bf = load_kfrag(Bs, k0, nhalf + j * 16, K1_PB);
        acc[j] = wmma_f32(af, bf, acc[j]);
      }
    }
    __syncthreads();   // all waves done reading before this buffer is restaged
    (void)curbuf;
  }

  // epilogue: C layout: VGPR r -> M = r + 8*(lane/16), N = lane%16
  const int half = lane >> 4;
  const int lc   = lane & 15;
#pragma unroll
  for (int j = 0; j < 4; ++j) {
    int col  = n0 + nhalf + j * 16 + lc;
    float bv = bias[col];
#pragma unroll
    for (int r = 0; r < 8; ++r) {
      int row = m0 + mstrip + r + 8 * half;
      Z[(size_t)row * GATES + col] = acc[j][r] + bv;
    }
  }
}

// ---------------------------------------------------------------------------
// Kernel 2: the scan. grid=(16 col-groups, 4 batch-tiles), 256 threads.
// WG (cg,bt): batch rows [16bt,16bt+16), hidden cols [32cg,32cg+32)
// => 512x128 Wh slice resident in LDS (TDM-loaded, 129-dword pitch).
// ---------------------------------------------------------------------------
#define K2_PW 129   // Wh slice pitch ([k][n], 512x128) -> 264192 B @ LDS 0
#define K2_PH 17    // h tile pitch   ([k][m], 512x16)  -> 34816 B
#define K2_PZ 132   // z slab pitch   ([m][n], 16x128)  -> 8448 B
#define K2_HS_OFF (512 * K2_PW * 4)                 // h_s LDS byte offset
#define K2_LDS_FLOATS (512 * K2_PW + 512 * K2_PH + 16 * K2_PZ)

__global__ __launch_bounds__(256) void lstm_scan(
    const float* __restrict__ Z, const float* __restrict__ Wh,
    const int* __restrict__ resets, const float* __restrict__ c0,
    const float* __restrict__ h0, float* __restrict__ ys,
    unsigned* __restrict__ bar) {
  extern __shared__ float smem[];
  float* Wh_s = smem;                        // [k][n] pitch K2_PW, LDS byte 0
  float* h_s  = Wh_s + 512 * K2_PW;          // [k][m] pitch K2_PH
  float* z_s  = h_s + 512 * K2_PH;           // [m][n] pitch K2_PZ

  const int tid  = threadIdx.x;
  const int wave = tid >> 5;
  const int lane = tid & 31;
  const int half = lane >> 4;
  const int lc   = lane & 15;
  const int cg   = blockIdx.x;       // 0..15 column group
  const int bt   = blockIdx.y;       // 0..3  batch tile
  const int b0   = bt * 16;
  const int c0h  = cg * 32;
  unsigned* cnt  = bar + bt;

  const int q    = wave >> 1;        // gate 0..3 (i,f,g,o)
  const int ccw  = (wave & 1) * 16;
  const int nloc = wave * 16;        // LDS slab column base (= q*32 + ccw)
  const int gcol = q * HID + c0h + ccw + lc;   // this lane's Zpre column

  // ---- one-time: Wh slice -> LDS. 4 TDM descriptors (one per gate strip).
#if USE_TDM
  if (wave == 0) {
    for (int qq = 0; qq < 4; ++qq)
      tdm_load_strip(Wh + (size_t)qq * HID + c0h, (unsigned)(qq * 32 * 4));
    __builtin_amdgcn_s_wait_tensorcnt(0);
  }
#else
  for (int idx = tid; idx < 512 * 128; idx += 256) {
    int k  = idx >> 7;
    int n  = idx & 127;
    Wh_s[k * K2_PW + n] = Wh[(size_t)k * GATES + (n >> 5) * HID + c0h + (n & 31)];
  }
#endif

  // ---- c state: 2 cells per thread, in registers
  float creg[2];
#pragma unroll
  for (int j = 0; j < 2; ++j) {
    int idx = tid + 256 * j;
    int m   = idx >> 5;
    int cc  = idx & 31;
    creg[j] = c0[(size_t)(b0 + m) * HID + c0h + cc];
  }
  __syncthreads();

  for (int t = 0; t < T_STEPS; ++t) {
    // ---- stage h (reset-masked) transposed into h_s[k][m] via async scatter.
    // 256 instructions/WG -> 32 per wave; wave w covers rows 2w, 2w+1.
    {
      const float* hsrc = (t == 0) ? (h0 + (size_t)b0 * HID)
                                   : (ys + ((size_t)(t - 1) * BATCH + b0) * HID);
      bool rst0 = (resets[t * BATCH + b0 + 2 * wave + 0] != 0);
      bool rst1 = (resets[t * BATCH + b0 + 2 * wave + 1] != 0);
#pragma unroll
      for (int i = 0; i < 32; ++i) {
        int m   = 2 * wave + (i & 1);
        int kc  = (i >> 1) * 32;
        bool rst = (i & 1) ? rst1 : rst0;    // wave-uniform branch
        if (!rst) {
          async_g2l_b32((unsigned)(K2_HS_OFF + ((kc + lane) * K2_PH + m) * 4),
                        hsrc + (size_t)m * HID + kc + lane);
        } else {
          h_s[(kc + lane) * K2_PH + m] = 0.f;
        }
      }
      wait_asynccnt0();
    }
    __syncthreads();

    // ---- Zpre slab into its own registers: overlaps the whole WMMA loop
    v8f zpre;
    {
      const float* zp = Z + (size_t)(t * BATCH + b0) * GATES + gcol;
#pragma unroll
      for (int r = 0; r < 8; ++r) zpre[r] = zp[(size_t)(r + 8 * half) * GATES];
    }
    // warm next step's Zpre slab in GL2 (counter-free prefetch)
    if (t + 1 < T_STEPS) {
      const float* zn = Z + (size_t)((t + 1) * BATCH + b0) * GATES + gcol;
#pragma unroll
      for (int r = 0; r < 8; ++r)
        __builtin_prefetch(zn + (size_t)(r + 8 * half) * GATES, 0, 3);
    }

    // ---- z-slab GEMM: 4 independent accumulation chains
    v8f acc0 = vzero8(), acc1 = vzero8(), acc2 = vzero8(), acc3 = vzero8();
    for (int k0 = 0; k0 < HID; k0 += 16) {
      v2f a0 = load_kfrag(h_s, k0 + 0, 0, K2_PH);
      v2f w0 = load_kfrag(Wh_s, k0 + 0, nloc, K2_PW);
      acc0 = wmma_f32(a0, w0, acc0);
      v2f a1 = load_kfrag(h_s, k0 + 4, 0, K2_PH);
      v2f w1 = load_kfrag(Wh_s, k0 + 4, nloc, K2_PW);
      acc1 = wmma_f32(a1, w1, acc1);
      v2f a2 = load_kfrag(h_s, k0 + 8, 0, K2_PH);
      v2f w2 = load_kfrag(Wh_s, k0 + 8, nloc, K2_PW);
      acc2 = wmma_f32(a2, w2, acc2);
      v2f a3 = load_kfrag(h_s, k0 + 12, 0, K2_PH);
      v2f w3 = load_kfrag(Wh_s, k0 + 12, nloc, K2_PW);
      acc3 = wmma_f32(a3, w3, acc3);
    }
#pragma unroll
    for (int r = 0; r < 8; ++r) {
      float zv = zpre[r] + acc0[r] + acc1[r] + acc2[r] + acc3[r];
      z_s[(r + 8 * half) * K2_PZ + nloc + lc] = zv;
    }
    __syncthreads();

    // ---- gates + state update + output (2 cells per thread)
#pragma unroll
    for (int j = 0; j < 2; ++j) {
      int idx = tid + 256 * j;
      int m   = idx >> 5;
      int cc  = idx & 31;
      bool rst = (resets[t * BATCH + b0 + m] != 0);
      float cv = rst ? 0.f : creg[j];
      const float* zr = z_s + m * K2_PZ;
      float iv = 1.f / (1.f + __expf(-zr[cc]));
      float fv = 1.f / (1.f + __expf(-zr[32 + cc]));
      float gv = tanhf(zr[64 + cc]);
      float ov = 1.f / (1.f + __expf(-zr[96 + cc]));
      float nc = fv * cv + iv * gv;            // pre-clip nc feeds nh
      float nh = ov * tanhf(nc);
      nc = fminf(1.f, fmaxf(-1.f, nc));        // straight-through clip (fwd)
      nh = fminf(1.f, fmaxf(-1.f, nh));
      creg[j] = nc;
      ys[((size_t)t * BATCH + b0 + m) * HID + c0h + cc] = nh;
    }

    // ---- per-step barrier across the 16 WGs of this batch tile
    __threadfence();
    __syncthreads();
    if (tid == 0) {
      __hip_atomic_fetch_add(cnt, 1u, __ATOMIC_RELEASE, __HIP_MEMORY_SCOPE_AGENT);
      unsigned target = 16u * (unsigned)(t + 1);   // monotonic: replay-safe
      while (__hip_atomic_load(cnt, __ATOMIC_ACQUIRE, __HIP_MEMORY_SCOPE_AGENT) < target) {
        __builtin_amdgcn_s_sleep(2);
      }
    }
    __syncthreads();
    __threadfence();   // acquire: drop stale near-cache lines before reading ys
  }
}

// ---------------------------------------------------------------------------
extern "C" void kernel_launch(void* const* d_in, const int* in_sizes, int n_in,
                              void* d_out, int out_size, void* d_ws, size_t ws_size,
                              hipStream_t stream) {
  const float* xs     = (const float*)d_in[0];
  const int*   resets = (const int*)d_in[1];
  const float* c0     = (const float*)d_in[2];
  const float* h0     = (const float*)d_in[3];
  const float* Wi     = (const float*)d_in[4];
  const float* Wh     = (const float*)d_in[5];
  const float* b      = (const float*)d_in[6];
  float* ys = (float*)d_out;

  float* Z = (float*)d_ws;                                   // 256 MB Zpre
  size_t zbytes = (size_t)T_STEPS * BATCH * GATES * sizeof(float);
  unsigned* bar = (unsigned*)((char*)d_ws + zbytes);         // 4 counters

  hipMemsetAsync(bar, 0, 4 * sizeof(unsigned), stream);      // replay-safe reset

  dim3 blk(256);
  dim3 g1((T_STEPS * BATCH) / 64, GATES / 128);              // 512 x 16
  hipLaunchKernelGGL(lstm_pregemm, g1, blk, (size_t)K1_LDS_BYTES, stream,
                     xs, Wi, b, Z);

  size_t lds2 = (size_t)K2_LDS_FLOATS * sizeof(float);       // ~300 KB WGP LDS
  hipFuncSetAttribute((const void*)lstm_scan,
                      hipFuncAttributeMaxDynamicSharedMemorySize, (int)lds2);
  dim3 g2(16, 4);
  hipLaunchKernelGGL(lstm_scan, g2, blk, lds2, stream,
                     Z, Wh, resets, c0, h0, ys, bar);
}